// PhiNN_18674517803038
// MI455X (gfx1250) — compile-verified
//
#include <hip/hip_runtime.h>
#include <stdint.h>

typedef float v2f __attribute__((ext_vector_type(2)));
typedef float v8f __attribute__((ext_vector_type(8)));

#define NDIMC   2
#define NCELLSC 200
#define BATCHC  128
#define XW      407                  // 2 + 400 + 5 columns of x
#define NPTS    (BATCHC * NCELLSC)   // 25600 points
#define NELEM   (NPTS * NDIMC)       // 51200 noise elements per step
#define HALFE   (NELEM / 2)          // 25600
#define DTC     0.001f
#define SIGMAC  0.001f
#define SQRT_DT 0.031622776601683794f
#define SQRT_2  1.4142135623730951f
#define LOG2E   1.4426950408889634f
#define LN2     0.6931471805599453f
#define LDP     33                   // padded LDS row stride (floats)

// ---------------- raw-hardware transcendentals ----------------
__device__ __forceinline__ float fast_log(float x) {   // ln(x)
  return __builtin_amdgcn_logf(x) * LN2;
}
__device__ __forceinline__ float fast_rcp(float x) {
  return __builtin_amdgcn_rcpf(x);
}

// ---------------- threefry2x32 (matches JAX) ----------------
__device__ __forceinline__ uint32_t rotl32(uint32_t v, int s) {
  return (v << s) | (v >> (32 - s));
}
__device__ __forceinline__ void threefry2x32(uint32_t k0, uint32_t k1,
                                             uint32_t x0, uint32_t x1,
                                             uint32_t& o0, uint32_t& o1) {
  uint32_t k2 = k0 ^ k1 ^ 0x1BD11BDAu;
  uint32_t y0 = x0 + k0, y1 = x1 + k1;
#define TFR(R) { y0 += y1; y1 = rotl32(y1, R); y1 ^= y0; }
  TFR(13) TFR(15) TFR(26) TFR(6)
  y0 += k1; y1 += k2 + 1u;
  TFR(17) TFR(29) TFR(16) TFR(24)
  y0 += k2; y1 += k0 + 2u;
  TFR(13) TFR(15) TFR(26) TFR(6)
  y0 += k0; y1 += k1 + 3u;
  TFR(17) TFR(29) TFR(16) TFR(24)
  y0 += k1; y1 += k2 + 4u;
  TFR(13) TFR(15) TFR(26) TFR(6)
  y0 += k2; y1 += k0 + 5u;
#undef TFR
  o0 = y0; o1 = y1;
}

// ---------------- erfinv (Giles, fp32) ----------------
__device__ __forceinline__ float erfinv_f(float x) {
  float w = -fast_log((1.0f - x) * (1.0f + x));
  float p;
  if (w < 5.0f) {
    w -= 2.5f;
    p = 2.81022636e-08f;
    p = fmaf(p, w, 3.43273939e-07f);
    p = fmaf(p, w, -3.5233877e-06f);
    p = fmaf(p, w, -4.39150654e-06f);
    p = fmaf(p, w, 0.00021858087f);
    p = fmaf(p, w, -0.00125372503f);
    p = fmaf(p, w, -0.00417768164f);
    p = fmaf(p, w, 0.246640727f);
    p = fmaf(p, w, 1.50140941f);
  } else {
    w = __builtin_amdgcn_sqrtf(w) - 3.0f;
    p = -0.000200214257f;
    p = fmaf(p, w, 0.000100950558f);
    p = fmaf(p, w, 0.00134934322f);
    p = fmaf(p, w, -0.00367342844f);
    p = fmaf(p, w, 0.00573950773f);
    p = fmaf(p, w, -0.0076224613f);
    p = fmaf(p, w, 0.00943887047f);
    p = fmaf(p, w, 1.00167406f);
    p = fmaf(p, w, 2.83297682f);
  }
  return p * x;
}

// jax.random.normal element e under fold_in(key, step) = (fk0, fk1)
__device__ __forceinline__ float jax_normal_elem(uint32_t fk0, uint32_t fk1, uint32_t e) {
  uint32_t c0, c1, o0, o1;
  bool first = (e < (uint32_t)HALFE);
  c0 = first ? e : (e - (uint32_t)HALFE);
  c1 = first ? (e + (uint32_t)HALFE) : e;
  threefry2x32(fk0, fk1, c0, c1, o0, o1);
  uint32_t bits = first ? o0 : o1;
  float u01 = __uint_as_float((bits >> 9) | 0x3f800000u) - 1.0f;   // [0,1)
  const float lo = -0.99999994f;                                   // nextafter(-1,0)
  float u = fmaf(u01, 1.0f - lo, lo);
  u = fmaxf(u, lo);
  return SQRT_2 * erfinv_f(u);
}

// ---------------- fused activations ----------------
// one exp shared between sigmoid and softplus:
//   e = exp(-|z|); t = 1/(1+e); sigmoid = z>=0 ? t : 1-t
//   softplus = max(z,0) + ln(1+e)
__device__ __forceinline__ void act_pair(float z, float& sig, float& sp) {
  float e = __builtin_amdgcn_exp2f(-fabsf(z) * LOG2E);
  float o = 1.0f + e;
  float t = fast_rcp(o);
  sig = (z >= 0.0f) ? t : 1.0f - t;
  sp  = fmaxf(z, 0.0f) + fast_log(o);
}
__device__ __forceinline__ float sigmoid_f(float z) {
  float e = __builtin_amdgcn_exp2f(-fabsf(z) * LOG2E);
  float t = fast_rcp(1.0f + e);
  return (z >= 0.0f) ? t : 1.0f - t;
}

// ---------------- WMMA f32 16x16x4 ----------------
__device__ __forceinline__ v8f wmma4(v2f a, v2f b, v8f c) {
  return __builtin_amdgcn_wmma_f32_16x16x4_f32(false, a, false, b, (short)0, c,
                                               false, false);
}

// B fragment for forward (B = W^T): B[k][n] = W[n*IN + k]
__device__ __forceinline__ v2f bfragT(const float* W, int IN, int k0, int n0, int lane) {
  int n  = n0 + (lane & 15);
  int kb = k0 + ((lane >> 4) << 1);
  v2f b; b.x = W[n * IN + kb]; b.y = W[n * IN + kb + 1]; return b;
}
// B fragment for backward (B = W): B[k][n] = W[k*C + n]
__device__ __forceinline__ v2f bfragN(const float* W, int C, int k0, int n0, int lane) {
  int n  = n0 + (lane & 15);
  int kb = k0 + ((lane >> 4) << 1);
  v2f b; b.x = W[kb * C + n]; b.y = W[(kb + 1) * C + n]; return b;
}
// A fragment (16xK chunk kc) from LDS tile (row-major, LDP stride)
__device__ __forceinline__ v2f afrag(const float* t, int kc, int lane) {
  int m  = lane & 15;
  int kb = 4 * kc + ((lane >> 4) << 1);
  v2f a; a.x = t[m * LDP + kb]; a.y = t[m * LDP + kb + 1]; return a;
}
// store a D-layout 16x16 tile (cols n0..n0+15) to LDS
__device__ __forceinline__ void dstore(float* t, v8f v, int n0, int lane) {
  int n  = n0 + (lane & 15);
  int mo = (lane >> 4) * 8;
#pragma unroll
  for (int r = 0; r < 8; ++r) t[(mo + r) * LDP + n] = v[r];
}

__global__ __launch_bounds__(32) void phi_sde_kernel(
    const float* __restrict__ w0, const float* __restrict__ b0,
    const float* __restrict__ w1, const float* __restrict__ b1,
    const float* __restrict__ w2, const float* __restrict__ b2,
    const float* __restrict__ w3, const float* __restrict__ b3,
    const float* __restrict__ w4, const float* __restrict__ b4,
    const float* __restrict__ wt, const float* __restrict__ x,
    float* __restrict__ out) {
  __shared__ float tile[16 * LDP];

  const int lane = (int)threadIdx.x;
  const int n    = lane & 15;
  const bool hi  = lane >= 16;

  const int nsteps = (int)rintf((x[1] - x[0]) / DTC);

  // Each of lanes 0..15 owns one point; lanes 16..31 mirror for indexing.
  const int p    = (int)blockIdx.x * 16 + n;
  const int bidx = p / NCELLSC;
  const int cidx = p - bidx * NCELLSC;
  const float* xb = x + bidx * XW;

  float yx = 0.0f, yy = 0.0f;
  if (!hi) { yx = xb[2 + cidx * 2]; yy = xb[2 + cidx * 2 + 1]; }
  const float tstart = xb[0];
  const float sw   = xb[XW - 5];           // sigp[:,0]  (switch time)
  const float sA0  = xb[XW - 4], sA1 = xb[XW - 3];
  const float sB0  = xb[XW - 2], sB1 = xb[XW - 1];
  const float wt00 = wt[0], wt01 = wt[1], wt10 = wt[2], wt11 = wt[3];

  // per-lane bias / head values (column = n in every D tile)
  const float bn0 = b0[n];
  const float bn1a = b1[n], bn1b = b1[16 + n];
  const float bn2a = b2[n], bn2b = b2[16 + n];
  const float bn3 = b3[n];
  const float w4n = w4[n];
  (void)b4;  // grad of sum kills b4

  // ---- weight fragments (hoisted, register-resident) ----
  v2f B0f;                                  // layer0 fwd: K padded 2->4
  { int kb = (lane >> 4) << 1;
    B0f.x = (kb == 0) ? w0[n * 2 + 0] : 0.0f;
    B0f.y = (kb == 0) ? w0[n * 2 + 1] : 0.0f; }
  v2f B1f[4][2], B2f[8][2], B3f[8];
#pragma unroll
  for (int kc = 0; kc < 4; ++kc)
#pragma unroll
    for (int nt = 0; nt < 2; ++nt) B1f[kc][nt] = bfragT(w1, 16, 4 * kc, 16 * nt, lane);
#pragma unroll
  for (int kc = 0; kc < 8; ++kc)
#pragma unroll
    for (int nt = 0; nt < 2; ++nt) B2f[kc][nt] = bfragT(w2, 32, 4 * kc, 16 * nt, lane);
#pragma unroll
  for (int kc = 0; kc < 8; ++kc) B3f[kc] = bfragT(w3, 32, 4 * kc, 0, lane);

  v2f W3b[4][2], W2b[8][2], W1b[8], W0b[4];
#pragma unroll
  for (int kc = 0; kc < 4; ++kc)
#pragma unroll
    for (int nt = 0; nt < 2; ++nt) W3b[kc][nt] = bfragN(w3, 32, 4 * kc, 16 * nt, lane);
#pragma unroll
  for (int kc = 0; kc < 8; ++kc)
#pragma unroll
    for (int nt = 0; nt < 2; ++nt) W2b[kc][nt] = bfragN(w2, 32, 4 * kc, 16 * nt, lane);
#pragma unroll
  for (int kc = 0; kc < 8; ++kc) W1b[kc] = bfragN(w1, 16, 4 * kc, 0, lane);
#pragma unroll
  for (int kc = 0; kc < 4; ++kc) {
    int kb = 4 * kc + ((lane >> 4) << 1);
    W0b[kc].x = (n < 2) ? w0[kb * 2 + n] : 0.0f;
    W0b[kc].y = (n < 2) ? w0[(kb + 1) * 2 + n] : 0.0f;
  }

  for (int i = 0; i < nsteps; ++i) {
    uint32_t fk0, fk1;
    threefry2x32(0u, 42u, 0u, (uint32_t)i, fk0, fk1);  // fold_in(key(42), i)

    // ================= forward =================
    v2f a0; a0.x = hi ? 0.0f : yx; a0.y = hi ? 0.0f : yy;
    v8f z0;
#pragma unroll
    for (int r = 0; r < 8; ++r) z0[r] = bn0;
    z0 = wmma4(a0, B0f, z0);
    v8f s0, sp;
#pragma unroll
    for (int r = 0; r < 8; ++r) {
      float sg, sv; act_pair(z0[r], sg, sv);
      s0[r] = sg; sp[r] = sv;
    }
    dstore(tile, sp, 0, lane);
    __syncthreads();
    v2f f1[4];
#pragma unroll
    for (int kc = 0; kc < 4; ++kc) f1[kc] = afrag(tile, kc, lane);
    __syncthreads();

    v8f s1[2];
#pragma unroll
    for (int nt = 0; nt < 2; ++nt) {
      v8f c;
      float bb = nt ? bn1b : bn1a;
#pragma unroll
      for (int r = 0; r < 8; ++r) c[r] = bb;
#pragma unroll
      for (int kc = 0; kc < 4; ++kc) c = wmma4(f1[kc], B1f[kc][nt], c);
      v8f spv;
#pragma unroll
      for (int r = 0; r < 8; ++r) {
        float sg, sv; act_pair(c[r], sg, sv);
        s1[nt][r] = sg; spv[r] = sv;
      }
      dstore(tile, spv, 16 * nt, lane);
    }
    __syncthreads();
    v2f f2[8];
#pragma unroll
    for (int kc = 0; kc < 8; ++kc) f2[kc] = afrag(tile, kc, lane);
    __syncthreads();

    v8f s2[2];
#pragma unroll
    for (int nt = 0; nt < 2; ++nt) {
      v8f c;
      float bb = nt ? bn2b : bn2a;
#pragma unroll
      for (int r = 0; r < 8; ++r) c[r] = bb;
#pragma unroll
      for (int kc = 0; kc < 8; ++kc) c = wmma4(f2[kc], B2f[kc][nt], c);
      v8f spv;
#pragma unroll
      for (int r = 0; r < 8; ++r) {
        float sg, sv; act_pair(c[r], sg, sv);
        s2[nt][r] = sg; spv[r] = sv;
      }
      dstore(tile, spv, 16 * nt, lane);
    }
    __syncthreads();
    v2f f3[8];
#pragma unroll
    for (int kc = 0; kc < 8; ++kc) f3[kc] = afrag(tile, kc, lane);
    __syncthreads();

    v8f z3;
#pragma unroll
    for (int r = 0; r < 8; ++r) z3[r] = bn3;
#pragma unroll
    for (int kc = 0; kc < 8; ++kc) z3 = wmma4(f3[kc], B3f[kc], z3);

    // ================= backward =================
    v8f d3;
#pragma unroll
    for (int r = 0; r < 8; ++r) d3[r] = w4n * sigmoid_f(z3[r]);
    dstore(tile, d3, 0, lane);
    __syncthreads();
    v2f g3[4];
#pragma unroll
    for (int kc = 0; kc < 4; ++kc) g3[kc] = afrag(tile, kc, lane);
    __syncthreads();

#pragma unroll
    for (int nt = 0; nt < 2; ++nt) {
      v8f c;
#pragma unroll
      for (int r = 0; r < 8; ++r) c[r] = 0.0f;
#pragma unroll
      for (int kc = 0; kc < 4; ++kc) c = wmma4(g3[kc], W3b[kc][nt], c);
      v8f d2;
#pragma unroll
      for (int r = 0; r < 8; ++r) d2[r] = c[r] * s2[nt][r];
      dstore(tile, d2, 16 * nt, lane);
    }
    __syncthreads();
    v2f g2[8];
#pragma unroll
    for (int kc = 0; kc < 8; ++kc) g2[kc] = afrag(tile, kc, lane);
    __syncthreads();

#pragma unroll
    for (int nt = 0; nt < 2; ++nt) {
      v8f c;
#pragma unroll
      for (int r = 0; r < 8; ++r) c[r] = 0.0f;
#pragma unroll
      for (int kc = 0; kc < 8; ++kc) c = wmma4(g2[kc], W2b[kc][nt], c);
      v8f d1;
#pragma unroll
      for (int r = 0; r < 8; ++r) d1[r] = c[r] * s1[nt][r];
      dstore(tile, d1, 16 * nt, lane);
    }
    __syncthreads();
    v2f g1[8];
#pragma unroll
    for (int kc = 0; kc < 8; ++kc) g1[kc] = afrag(tile, kc, lane);
    __syncthreads();

    v8f c0;
#pragma unroll
    for (int r = 0; r < 8; ++r) c0[r] = 0.0f;
#pragma unroll
    for (int kc = 0; kc < 8; ++kc) c0 = wmma4(g1[kc], W1b[kc], c0);
    v8f d0;
#pragma unroll
    for (int r = 0; r < 8; ++r) d0[r] = c0[r] * s0[r];
    dstore(tile, d0, 0, lane);
    __syncthreads();
    v2f g0[4];
#pragma unroll
    for (int kc = 0; kc < 4; ++kc) g0[kc] = afrag(tile, kc, lane);
    __syncthreads();

    v8f gr;
#pragma unroll
    for (int r = 0; r < 8; ++r) gr[r] = 0.0f;
#pragma unroll
    for (int kc = 0; kc < 4; ++kc) gr = wmma4(g0[kc], W0b[kc], gr);

    // grad tile: only cols 0,1 meaningful
    if (n < 2) {
      int mo = (lane >> 4) * 8;
#pragma unroll
      for (int r = 0; r < 8; ++r) tile[(mo + r) * LDP + n] = gr[r];
    }
    __syncthreads();

    if (!hi) {
      float gx = tile[lane * LDP + 0];
      float gy = tile[lane * LDP + 1];
      float ts = tstart + (float)i * DTC;
      bool pre = ts < sw;
      float sv0 = pre ? sA0 : sB0;
      float sv1 = pre ? sA1 : sB1;
      float tx = sv0 * wt00 + sv1 * wt01;
      float ty = sv0 * wt10 + sv1 * wt11;
      uint32_t e0 = (uint32_t)(2 * p);
      float nx = jax_normal_elem(fk0, fk1, e0);
      float ny = jax_normal_elem(fk0, fk1, e0 + 1u);
      yx += -(gx + tx) * DTC + SIGMAC * SQRT_DT * nx;
      yy += -(gy + ty) * DTC + SIGMAC * SQRT_DT * ny;
    }
    __syncthreads();
  }

  if (!hi) {
    out[2 * p]     = yx;
    out[2 * p + 1] = yy;
  }
}

extern "C" void kernel_launch(void* const* d_in, const int* in_sizes, int n_in,
                              void* d_out, int out_size, void* d_ws, size_t ws_size,
                              hipStream_t stream) {
  (void)in_sizes; (void)n_in; (void)out_size; (void)d_ws; (void)ws_size;
  // setup_inputs() dict order: w0,b0,w1,b1,w2,b2,w3,b3,w4,b4,wt,x
  const float* w0 = (const float*)d_in[0];
  const float* b0 = (const float*)d_in[1];
  const float* w1 = (const float*)d_in[2];
  const float* b1 = (const float*)d_in[3];
  const float* w2 = (const float*)d_in[4];
  const float* b2 = (const float*)d_in[5];
  const float* w3 = (const float*)d_in[6];
  const float* b3 = (const float*)d_in[7];
  const float* w4 = (const float*)d_in[8];
  const float* b4 = (const float*)d_in[9];
  const float* wt = (const float*)d_in[10];
  const float* x  = (const float*)d_in[11];
  float* out = (float*)d_out;

  dim3 grid(NPTS / 16);  // 1600 waves, one 16-point tile each
  dim3 block(32);        // single wave per workgroup -> barriers are free
  phi_sde_kernel<<<grid, block, 0, stream>>>(w0, b0, w1, b1, w2, b2, w3, b3,
                                             w4, b4, wt, x, out);
}